// CM_SVMHead_36412732735941
// MI455X (gfx1250) — compile-verified
//
#include <hip/hip_runtime.h>
#include <hip/hip_bf16.h>
#include <cstddef>
#include <cstdint>

// ---------------------------------------------------------------------------
// MetaOptNet SVM head on MI455X (gfx1250).
//   Kernel 1 (WMMA f32): K[t] = S[t] S[t]^T           (64 x [75x4096x75])
//   Kernel 2 (TDM+LDS QP): 15-iter interior point via Schur complement:
//                        five 75x75 SPD inversions + one 75x75 inversion
//                        per iteration instead of a 450x450 dense solve.
//                        K tile staged into LDS with TENSOR_LOAD_TO_LDS.
//   Kernel 3 (VALU):     W[t] = S[t]^T Z[t]           (4096x5 per task)
//   Kernel 4 (WMMA f32): logits[t] = Q[t] W[t]        (150x4096x5)
// ---------------------------------------------------------------------------

typedef __attribute__((ext_vector_type(2))) float v2f;
typedef __attribute__((ext_vector_type(8))) float v8f;
typedef __attribute__((ext_vector_type(4))) unsigned int u32x4;
typedef __attribute__((ext_vector_type(8))) int i32x8;
typedef __attribute__((ext_vector_type(4))) int i32x4;

#define TASKS 64
#define NSUP  75
#define NWAY  5
#define NV    375          // NSUP * NWAY
#define NQ    150
#define DIM   4096
#define KPAD  80           // padded Gram stride (5 x 16 WMMA tiles)
#define ST    76           // LDS / global row stride for work matrices
#define ITERS 15
#define NTH   256
#define CREG  0.1f
#define SIG   0.1f

// ---------------------------------------------------------------------------
// Kernel 1: Gram matrix K = S S^T with V_WMMA_F32_16X16X4_F32.
// One wave per 16x16 output tile; 5x5 tiles per task (75 padded to 80).
// For S*S^T the A and B fragments have identical lane indexing:
//   lane l holds row (tile*16 + (l&15)), K = k0 + 2*(l>>4) + {0,1}  (float2).
// ---------------------------------------------------------------------------
__global__ __launch_bounds__(32) void gram_kernel(
    const float* __restrict__ support, float* __restrict__ Kp) {
  const int task = blockIdx.y;
  const int tm = blockIdx.x / 5, tn = blockIdx.x % 5;
  const int lane = threadIdx.x;
  const int ln = lane & 15, hi = lane >> 4;

  int rowA = tm * 16 + ln; if (rowA > NSUP - 1) rowA = NSUP - 1; // pad rows: clamp (output ignored)
  int rowB = tn * 16 + ln; if (rowB > NSUP - 1) rowB = NSUP - 1;

  const float* pa = support + ((size_t)task * NSUP + rowA) * DIM + 2 * hi;
  const float* pb = support + ((size_t)task * NSUP + rowB) * DIM + 2 * hi;

  v8f acc = {};
#pragma unroll 8
  for (int k0 = 0; k0 < DIM; k0 += 4) {
    v2f a = *(const v2f*)(pa + k0);
    v2f b = *(const v2f*)(pb + k0);
    acc = __builtin_amdgcn_wmma_f32_16x16x4_f32(
        false, a, false, b, (short)0, acc, false, false);
  }

  float* out = Kp + (size_t)task * KPAD * KPAD + (tm * 16) * KPAD + tn * 16;
#pragma unroll
  for (int vr = 0; vr < 8; ++vr)
    out[(vr + 8 * hi) * KPAD + ln] = acc[vr];  // D reg vr: M = vr + 8*hi, N = ln
}

// ---------------------------------------------------------------------------
// In-place Jordan inversion of an SPD 75x75 matrix in LDS (no pivoting).
// 3 barriers per elimination step.
// ---------------------------------------------------------------------------
__device__ inline void jordan_inv(float* M, float* fcol, int tid) {
  for (int k = 0; k < NSUP; ++k) {
    // gather old column k
    for (int j = tid; j < NSUP; j += NTH) fcol[j] = M[j * ST + k];
    __syncthreads();
    const float ip = 1.0f / fcol[k];
    // scale row k (diag becomes 1/pivot)
    for (int j = tid; j < NSUP; j += NTH)
      M[k * ST + j] = (j == k) ? ip : M[k * ST + j] * ip;
    __syncthreads();
    // eliminate all other rows
    for (int t = tid; t < NSUP * NSUP; t += NTH) {
      const int i = t / NSUP, j = t % NSUP;
      if (i == k) continue;
      const float f = fcol[i];
      M[i * ST + j] = (j == k) ? (-f * ip) : (M[i * ST + j] - f * M[k * ST + j]);
    }
    __syncthreads();
  }
}

// ---------------------------------------------------------------------------
// Kernel 2: QP interior-point solver. One workgroup (8 waves) per task.
// KKT system [[G+D, A^T],[A,0]] is solved via the class-block structure:
//   H_w = K + I + diag(D[:,w])  (SPD 75x75, w = 0..4)
//   S   = sum_w H_w^{-1};  S dnu = sum_w H_w^{-1} r1_w + ra
//   dz_w = H_w^{-1} r1_w - H_w^{-1} dnu
// K (75x75 tile of the 80-stride Gram tensor) is DMA'd into LDS by the
// Tensor Data Mover (TENSOR_LOAD_TO_LDS, wave 0 only, TENSORcnt-tracked),
// overlapping with state initialization. H_w^{-1} are staged through global
// scratch (L2-resident) for the dz pass.
// ---------------------------------------------------------------------------
__global__ __launch_bounds__(NTH) void qp_kernel(
    const float* __restrict__ Kp, const int* __restrict__ labels,
    float* __restrict__ Hg, float* __restrict__ qp_out) {
  __shared__ float Kl[NSUP * NSUP];       // 22.5 KB, stride 75 (odd -> no bank conflicts)
  __shared__ float Hbuf[NSUP * ST];
  __shared__ float Sbuf[NSUP * ST];
  __shared__ float zv[NV + 1], sv[NV + 1], lv[NV + 1];
  __shared__ float rsv[NV + 1], r1v[NV + 1], dzv[NV + 1], Uv[NV + 1];
  __shared__ float nuv[NSUP], rav[NSUP], gv[NSUP], dnuv[NSUP];
  __shared__ float fcol[NSUP];
  __shared__ float red[NTH];
  __shared__ int lbl[NSUP];

  const int task = blockIdx.x;
  const int tid = threadIdx.x;
  float* Hgt = Hg + (size_t)task * NWAY * NSUP * ST;

  // --- TDM: DMA the 75x75 K tile (80-stride tensor) into LDS -------------
  if (tid < 32) {  // one wave issues; TDM is per-wave, EXEC-ignored
    const unsigned lds_off = (unsigned)(uintptr_t)(&Kl[0]);  // LDS aperture: addr[31:0] == LDS byte offset
    const unsigned long long ga =
        (unsigned long long)(uintptr_t)(Kp + (size_t)task * KPAD * KPAD);
    u32x4 g0;
    g0[0] = 1u;                                         // count=1 (valid user D#)
    g0[1] = lds_off;                                    // lds_addr
    g0[2] = (unsigned)(ga & 0xffffffffu);               // global_addr[31:0]
    g0[3] = (unsigned)((ga >> 32) & 0x01ffffffu)        // global_addr[56:32]
          | (2u << 30);                                 // type=2 ("image")
    i32x8 g1;
    g1[0] = (int)(2u << 16);    // workgroup_mask=0, data_size=2 (4 bytes)
    g1[1] = (int)(80u << 16);   // tensor_dim0 = 80  (bits [63:48] = low 16)
    g1[2] = (int)(80u << 16);   // tensor_dim1 = 80  (bits [95:80] = low 16)
    g1[3] = (int)(75u << 16);   // tile_dim0  = 75   (bits [127:112])
    g1[4] = 75;                 // tile_dim1  = 75   (bits [143:128]); tile_dim2=0
    g1[5] = 80;                 // tensor_dim0_stride = 80 (bits [191:160])
    g1[6] = 0;                  // tensor_dim0_stride hi / tensor_dim1_stride lo
    g1[7] = 0;
    i32x4 gz4 = {0, 0, 0, 0};   // groups 2/3 unused (2D tensor)
    i32x8 gz8 = {0, 0, 0, 0, 0, 0, 0, 0};
    __builtin_amdgcn_tensor_load_to_lds(g0, g1, gz4, gz4, gz8, 0);
  }

  // --- state init (overlaps with the DMA) ---------------------------------
  for (int i = tid; i < NSUP; i += NTH) { lbl[i] = labels[task * NSUP + i]; nuv[i] = 0.f; }
  for (int v = tid; v < NV; v += NTH) { zv[v] = 0.f; sv[v] = 1.f; lv[v] = 1.f; }

  if (tid < 32) __builtin_amdgcn_s_wait_tensorcnt(0);
  __syncthreads();

  for (int it = 0; it < ITERS; ++it) {
    // --- residuals: rz (stashed in r1v), rs, ra ---------------------------
    for (int v = tid; v < NV; v += NTH) {
      const int i = v / NWAY, w = v % NWAY;
      const float* Krow = Kl + i * NSUP;
      float acc = 0.f;
      for (int j = 0; j < NSUP; ++j) acc += Krow[j] * zv[j * NWAY + w];
      const float y = (lbl[i] == w) ? 1.f : 0.f;
      r1v[v] = acc + zv[v] - y + lv[v] + nuv[i];   // rz
      rsv[v] = zv[v] + sv[v] - CREG * y;           // rs
    }
    for (int i = tid; i < NSUP; i += NTH) {
      float a = 0.f;
      for (int w = 0; w < NWAY; ++w) a += zv[i * NWAY + w];
      rav[i] = a;                                  // ra  (b == 0)
    }
    // --- mu = mean(lam * s) ----------------------------------------------
    float lmu = 0.f;
    for (int v = tid; v < NV; v += NTH) lmu += lv[v] * sv[v];
    red[tid] = lmu; __syncthreads();
    for (int off = NTH / 2; off > 0; off >>= 1) {
      if (tid < off) red[tid] += red[tid + off];
      __syncthreads();
    }
    const float mu = red[0] / (float)NV;

    // --- r1 = -rz - (lam*rs - lam*s + sigma*mu)/s ------------------------
    for (int v = tid; v < NV; v += NTH) {
      const float rz = r1v[v];
      r1v[v] = -rz - (lv[v] * rsv[v] - lv[v] * sv[v] + SIG * mu) / sv[v];
    }
    for (int t = tid; t < NSUP * ST; t += NTH) Sbuf[t] = 0.f;
    __syncthreads();

    // --- per-class inversions: S += H_w^{-1}, u_w = H_w^{-1} r1_w --------
    for (int w = 0; w < NWAY; ++w) {
      for (int t = tid; t < NSUP * NSUP; t += NTH) {
        const int i = t / NSUP, j = t % NSUP;
        float hv = Kl[i * NSUP + j];
        if (i == j) hv += 1.f + lv[i * NWAY + w] / sv[i * NWAY + w];
        Hbuf[i * ST + j] = hv;
      }
      __syncthreads();
      jordan_inv(Hbuf, fcol, tid);
      for (int t = tid; t < NSUP * NSUP; t += NTH) {
        const int i = t / NSUP, j = t % NSUP;
        const float hv = Hbuf[i * ST + j];
        Sbuf[i * ST + j] += hv;
        Hgt[(w * NSUP + i) * ST + j] = hv;         // stage inverse for dz pass
      }
      for (int i = tid; i < NSUP; i += NTH) {
        float a = 0.f;
        for (int j = 0; j < NSUP; ++j) a += Hbuf[i * ST + j] * r1v[j * NWAY + w];
        Uv[i * NWAY + w] = a;
      }
      __threadfence_block();
      __syncthreads();
    }

    // --- g = sum_w u_w + ra; invert S; dnu = S^{-1} g --------------------
    for (int i = tid; i < NSUP; i += NTH) {
      float a = rav[i];
      for (int w = 0; w < NWAY; ++w) a += Uv[i * NWAY + w];
      gv[i] = a;
    }
    __syncthreads();
    jordan_inv(Sbuf, fcol, tid);
    for (int i = tid; i < NSUP; i += NTH) {
      float a = 0.f;
      for (int j = 0; j < NSUP; ++j) a += Sbuf[i * ST + j] * gv[j];
      dnuv[i] = a;
    }
    __syncthreads();

    // --- dz_w = u_w - H_w^{-1} dnu (inverses read back through L2) -------
    for (int v = tid; v < NV; v += NTH) {
      const int i = v / NWAY, w = v % NWAY;
      const float* Hrow = Hgt + (w * NSUP + i) * ST;
      float a = 0.f;
      for (int j = 0; j < NSUP; ++j) a += Hrow[j] * dnuv[j];
      dzv[v] = Uv[v] - a;
    }

    // --- fraction-to-boundary step length --------------------------------
    float lmin = 1e30f;
    for (int v = tid; v < NV; v += NTH) {
      const float ds = -rsv[v] - dzv[v];
      const float dl = (lv[v] * dzv[v] + lv[v] * rsv[v] - lv[v] * sv[v] + SIG * mu) / sv[v];
      if (ds < 0.f) { const float r = -sv[v] / ds; if (r < lmin) lmin = r; }
      if (dl < 0.f) { const float r = -lv[v] / dl; if (r < lmin) lmin = r; }
    }
    red[tid] = lmin; __syncthreads();
    for (int off = NTH / 2; off > 0; off >>= 1) {
      if (tid < off) red[tid] = fminf(red[tid], red[tid + off]);
      __syncthreads();
    }
    const float alpha = fminf(1.f, 0.99f * red[0]);

    // --- update state -----------------------------------------------------
    for (int v = tid; v < NV; v += NTH) {
      const float ds = -rsv[v] - dzv[v];
      const float dl = (lv[v] * dzv[v] + lv[v] * rsv[v] - lv[v] * sv[v] + SIG * mu) / sv[v];
      zv[v] += alpha * dzv[v];
      sv[v] += alpha * ds;
      lv[v] += alpha * dl;
    }
    for (int i = tid; i < NSUP; i += NTH) nuv[i] += alpha * dnuv[i];
    __syncthreads();
  }

  for (int v = tid; v < NV; v += NTH)
    qp_out[(size_t)task * NV + v] = zv[v];
}

// ---------------------------------------------------------------------------
// Kernel 3: W[t] = S[t]^T Z[t]  (4096 x 5 per task). Coalesced over d.
// ---------------------------------------------------------------------------
__global__ __launch_bounds__(256) void wvec_kernel(
    const float* __restrict__ support, const float* __restrict__ qp,
    float* __restrict__ Wc) {
  __shared__ float q[NV];
  const int task = blockIdx.y;
  const int d = blockIdx.x * 256 + threadIdx.x;
  for (int v = threadIdx.x; v < NV; v += 256) q[v] = qp[(size_t)task * NV + v];
  __syncthreads();
  float acc[NWAY] = {0.f, 0.f, 0.f, 0.f, 0.f};
  const float* sp = support + (size_t)task * NSUP * DIM + d;
  for (int n = 0; n < NSUP; ++n) {
    const float s = sp[(size_t)n * DIM];
#pragma unroll
    for (int w = 0; w < NWAY; ++w) acc[w] += s * q[n * NWAY + w];
  }
  float* o = Wc + ((size_t)task * DIM + d) * NWAY;
#pragma unroll
  for (int w = 0; w < NWAY; ++w) o[w] = acc[w];
}

// ---------------------------------------------------------------------------
// Kernel 4: logits = Q W  via WMMA f32 (150 padded to 10 tiles of 16, N=5
// padded to 16 in the fragment; pad lanes contribute zeros).
// ---------------------------------------------------------------------------
__global__ __launch_bounds__(32) void logits_kernel(
    const float* __restrict__ query, const float* __restrict__ Wc,
    float* __restrict__ out) {
  const int task = blockIdx.y;
  const int tq = blockIdx.x;
  const int lane = threadIdx.x;
  const int ln = lane & 15, hi = lane >> 4;

  int row = tq * 16 + ln; if (row > NQ - 1) row = NQ - 1;
  const float* pq = query + ((size_t)task * NQ + row) * DIM + 2 * hi;
  const float* pw = Wc + (size_t)task * DIM * NWAY;
  const bool nvalid = (ln < NWAY);

  v8f acc = {};
#pragma unroll 4
  for (int k0 = 0; k0 < DIM; k0 += 4) {
    v2f a = *(const v2f*)(pq + k0);
    const int k = k0 + 2 * hi;
    v2f b;
    b.x = nvalid ? pw[(size_t)k * NWAY + ln] : 0.f;
    b.y = nvalid ? pw[(size_t)(k + 1) * NWAY + ln] : 0.f;
    acc = __builtin_amdgcn_wmma_f32_16x16x4_f32(
        false, a, false, b, (short)0, acc, false, false);
  }

#pragma unroll
  for (int vr = 0; vr < 8; ++vr) {
    const int m = tq * 16 + vr + 8 * hi;
    if (m < NQ && ln < NWAY)
      out[((size_t)task * NQ + m) * NWAY + ln] = acc[vr];
  }
}

// ---------------------------------------------------------------------------
// Host-side launcher. Workspace layout (floats):
//   Kp : TASKS*80*80            =   409,600
//   qp : TASKS*384 (375 used)   =    24,576
//   Wc : TASKS*4096*5           = 1,310,720
//   Hg : TASKS*5*75*76          = 1,824,000
// total ~14.3 MB of d_ws.
// ---------------------------------------------------------------------------
extern "C" void kernel_launch(void* const* d_in, const int* in_sizes, int n_in,
                              void* d_out, int out_size, void* d_ws, size_t ws_size,
                              hipStream_t stream) {
  const float* query   = (const float*)d_in[0];
  const float* support = (const float*)d_in[1];
  const int*   labels  = (const int*)d_in[2];

  float* ws = (float*)d_ws;
  float* Kp = ws;
  float* qp = Kp + (size_t)TASKS * KPAD * KPAD;
  float* Wc = qp + (size_t)TASKS * 384;
  float* Hg = Wc + (size_t)TASKS * DIM * NWAY;

  gram_kernel<<<dim3(25, TASKS), 32, 0, stream>>>(support, Kp);
  qp_kernel<<<dim3(TASKS), NTH, 0, stream>>>(Kp, labels, Hg, qp);
  wvec_kernel<<<dim3(DIM / 256, TASKS), 256, 0, stream>>>(support, qp, Wc);
  logits_kernel<<<dim3(10, TASKS), 32, 0, stream>>>(query, Wc, (float*)d_out);
}